// GraphEncoder_85392539779511
// MI455X (gfx1250) — compile-verified
//
#include <hip/hip_runtime.h>
#include <hip/hip_bf16.h>
#include <math.h>

#define F_IN   64
#define HID    64
#define EMB    32
#define NGRAPH 128
#define BN_EPS 1e-5f

typedef float v2f __attribute__((ext_vector_type(2)));
typedef float v4f __attribute__((ext_vector_type(4)));
typedef float v8f __attribute__((ext_vector_type(8)));

// ---------------------------------------------------------------- utilities
__global__ void fill_kernel(float* __restrict__ p, float v, long long n) {
  long long i = (long long)blockIdx.x * blockDim.x + threadIdx.x;
  if (i < n) p[i] = v;
}

// vectorized zero-fill: n4 = count of float4 elements (b128 stores)
__global__ void fill4_kernel(v4f* __restrict__ p, float v, long long n4) {
  long long i = (long long)blockIdx.x * blockDim.x + threadIdx.x;
  if (i < n4) { v4f q = {v, v, v, v}; p[i] = q; }
}

__device__ __forceinline__ void atomicMaxF(float* addr, float val) {
  unsigned int* ia = (unsigned int*)addr;
  unsigned int old = *ia;
  while (__uint_as_float(old) < val) {
    unsigned int assumed = old;
    old = atomicCAS(ia, assumed, __float_as_uint(val));
    if (old == assumed) break;
  }
}

// ---------------------------------------------------------------- gcn_norm
__global__ void deg_accum_kernel(const int* __restrict__ dst, float* __restrict__ deg, int e) {
  int i = blockIdx.x * blockDim.x + threadIdx.x;
  if (i < e) atomicAdd(&deg[dst[i]], 1.0f);
}

__global__ void dinv_kernel(float* __restrict__ dinv, int n) {
  int i = blockIdx.x * blockDim.x + threadIdx.x;
  if (i < n) dinv[i] = rsqrtf(fmaxf(dinv[i], 1.0f));
}

// ---------------------------------------------------------------- FP32 WMMA GEMM
// C[M,NC] = A[M,64] * B[64,NC]. Requires M % 16 == 0 (N = 50000 = 3125*16).
// NC compile-time (64 or 32) so all B/C offsets fold into immediate-offset
// vmem instructions. Each wave owns one 16-wide N tile and FOUR consecutive
// 16-row M tiles: the 16 B fragments (32 VGPRs) are loaded ONCE and reused
// across 4 tiles -> 64 x V_WMMA_F32_16X16X4_F32 per wave. No per-row store
// guards (M is tile-aligned), so C stores issue unguarded.
// f32 16x16x4 layouts: A: lanes 0-15 M=lane, K={0,1}+k0; lanes 16-31 K={2,3}+k0.
//                      B: lanes 0-15 N=lane, K={0,1}+k0; lanes 16-31 K={2,3}+k0.
//                      C: vgpr r, lanes 0-15 -> (M=r, N=lane); lanes 16-31 -> M=8+r.
template <int NC>
__global__ void gemm_wmma_f32(const float* __restrict__ A,
                              const float* __restrict__ B,
                              float* __restrict__ C,
                              int M) {
  constexpr int K = 64;
  const int lane = threadIdx.x & 31;
  const int wave = threadIdx.x >> 5;
  const int group = blockIdx.x * (blockDim.x >> 5) + wave;  // 1 N-tile x 4 M-tiles
  const int mGroups = (M + 63) >> 6;
  const int tileN = group / mGroups;
  const int mg    = group % mGroups;
  if (tileN >= NC / 16) return;              // wave-uniform: EXEC all-ones for WMMA

  const int half = lane >> 4;                // 0: K pair {0,1}, 1: K pair {2,3}
  const int l16  = lane & 15;
  const int nTiles = (M + 15) >> 4;

  // ---- preload all B fragments for this N tile (compile-time offsets)
  v2f bfrag[16];
  const float* __restrict__ bp = B + (2 * half) * NC + tileN * 16 + l16;
#pragma unroll
  for (int k = 0; k < 16; ++k) {
    bfrag[k].x = bp[(4 * k + 0) * NC];
    bfrag[k].y = bp[(4 * k + 1) * NC];
  }

  // ---- 4 M tiles reuse the resident B fragments
#pragma unroll
  for (int t = 0; t < 4; ++t) {
    const int tileM = mg * 4 + t;
    if (tileM >= nTiles) break;              // uniform across wave

    const float* __restrict__ ap = A + (size_t)(tileM * 16 + l16) * K + 2 * half;

    v8f c = {0.f, 0.f, 0.f, 0.f, 0.f, 0.f, 0.f, 0.f};
#pragma unroll
    for (int k = 0; k < 16; ++k) {
      v2f a = *(const v2f*)(ap + 4 * k);     // contiguous pair -> global_load_b64
      c = __builtin_amdgcn_wmma_f32_16x16x4_f32(false, a, false, bfrag[k],
                                                (short)0, c, false, false);
    }

    float* __restrict__ cp = C + (size_t)(tileM * 16 + half * 8) * NC + tileN * 16 + l16;
#pragma unroll
    for (int r = 0; r < 8; ++r)
      cp[r * NC] = c[r];                     // unguarded: M is 16-aligned
  }
}

// ---------------------------------------------------------------- edge scatter
// One wave32 per edge: src/dst/norm fetched once, features striped over lanes.
// agg (12.8 MB) is L2-resident (192 MB L2) -> fp32 add atomics resolve in L2.
template <int F>
__global__ void scatter_kernel(const float* __restrict__ hw,
                               const int* __restrict__ src,
                               const int* __restrict__ dst,
                               const float* __restrict__ dinv,
                               float* __restrict__ agg,
                               int e) {
  const int edge = blockIdx.x * (blockDim.x >> 5) + (threadIdx.x >> 5);
  const int lane = threadIdx.x & 31;
  if (edge >= e) return;
  const int s = src[edge];
  const int d = dst[edge];
  const float nrm = dinv[s] * dinv[d];
  const float* __restrict__ hs = hw + (size_t)s * F;
  float* __restrict__ ad = agg + (size_t)d * F;
  if (F == 64) {                             // b64 gather, 2 atomics per lane
    v2f v = *(const v2f*)(hs + 2 * lane);
    atomicAdd(&ad[2 * lane + 0], v.x * nrm);
    atomicAdd(&ad[2 * lane + 1], v.y * nrm);
  } else {                                   // F == 32: 1 float per lane
    atomicAdd(&ad[lane], hs[lane] * nrm);
  }
}

// ---------------------------------------------------------------- epilogues
// self-loop (norm = dinv^2) + bias + BN(eval) + ReLU, in place on agg.
// float4 per thread -> b128 load/store on the 12.8 MB streams.
__global__ void bn_relu_kernel(float* __restrict__ agg,
                               const float* __restrict__ hw,
                               const float* __restrict__ dinv,
                               const float* __restrict__ bias,
                               const float* __restrict__ gamma,
                               const float* __restrict__ beta,
                               const float* __restrict__ mean,
                               const float* __restrict__ var,
                               int n) {
  long long i = (long long)blockIdx.x * blockDim.x + threadIdx.x;   // n*16 quads
  if (i >= (long long)n * (HID / 4)) return;
  const int node = (int)(i >> 4);
  const int f4 = ((int)i & 15) * 4;
  const float di = dinv[node];
  const float d2 = di * di;
  v4f a = *(const v4f*)(agg + i * 4);
  v4f h = *(const v4f*)(hw + i * 4);
  v4f bi = *(const v4f*)(bias + f4);
  v4f mu = *(const v4f*)(mean + f4);
  v4f vr = *(const v4f*)(var + f4);
  v4f ga = *(const v4f*)(gamma + f4);
  v4f be = *(const v4f*)(beta + f4);
  v4f r;
#pragma unroll
  for (int j = 0; j < 4; ++j) {
    float v = a[j] + h[j] * d2 + bi[j];
    v = (v - mu[j]) * rsqrtf(vr[j] + BN_EPS) * ga[j] + be[j];
    r[j] = fmaxf(v, 0.0f);
  }
  *(v4f*)(agg + i * 4) = r;
}

// self-loop + bias only (final conv), F = EMB, float4 vectorized
__global__ void selfloop_bias_kernel(float* __restrict__ agg,
                                     const float* __restrict__ hw,
                                     const float* __restrict__ dinv,
                                     const float* __restrict__ bias,
                                     int n) {
  long long i = (long long)blockIdx.x * blockDim.x + threadIdx.x;   // n*8 quads
  if (i >= (long long)n * (EMB / 4)) return;
  const int node = (int)(i >> 3);
  const int f4 = ((int)i & 7) * 4;
  const float di = dinv[node];
  const float d2 = di * di;
  v4f a = *(const v4f*)(agg + i * 4);
  v4f h = *(const v4f*)(hw + i * 4);
  v4f bi = *(const v4f*)(bias + f4);
  v4f r;
#pragma unroll
  for (int j = 0; j < 4; ++j) r[j] = a[j] + h[j] * d2 + bi[j];
  *(v4f*)(agg + i * 4) = r;
}

// ---------------------------------------------------------------- pooling
__global__ void out_init_kernel(float* __restrict__ out) {
  int i = blockIdx.x * blockDim.x + threadIdx.x;
  if (i < NGRAPH * 2 * EMB) {
    const int f = i & (2 * EMB - 1);
    out[i] = (f < EMB) ? 0.0f : -__builtin_huge_valf();   // segment_max identity = -inf
  }
}

__global__ void count_kernel(const int* __restrict__ batch, float* __restrict__ counts, int n) {
  int i = blockIdx.x * blockDim.x + threadIdx.x;
  if (i < n) atomicAdd(&counts[batch[i]], 1.0f);
}

__global__ void pool_kernel(const float* __restrict__ h,
                            const int* __restrict__ batch,
                            float* __restrict__ out, int n) {
  long long i = (long long)blockIdx.x * blockDim.x + threadIdx.x;
  if (i >= (long long)n * EMB) return;
  const int node = (int)(i >> 5);
  const int f = (int)(i & 31);
  const int g = batch[node];
  const float v = h[i];
  atomicAdd(&out[g * (2 * EMB) + f], v);
  atomicMaxF(&out[g * (2 * EMB) + EMB + f], v);
}

__global__ void finalize_kernel(float* __restrict__ out, const float* __restrict__ counts) {
  int i = blockIdx.x * blockDim.x + threadIdx.x;
  if (i >= NGRAPH * EMB) return;
  const int g = i >> 5;
  const int f = i & 31;
  out[g * (2 * EMB) + f] /= fmaxf(counts[g], 1.0f);
}

// ---------------------------------------------------------------- launcher
extern "C" void kernel_launch(void* const* d_in, const int* in_sizes, int n_in,
                              void* d_out, int out_size, void* d_ws, size_t ws_size,
                              hipStream_t stream) {
  const float* x      = (const float*)d_in[0];
  const int*   ei     = (const int*)d_in[1];
  const int*   batch  = (const int*)d_in[2];
  const float* W1     = (const float*)d_in[3];
  const float* b1     = (const float*)d_in[4];
  const float* W2     = (const float*)d_in[5];
  const float* b2     = (const float*)d_in[6];
  const float* W3     = (const float*)d_in[7];
  const float* b3     = (const float*)d_in[8];
  const float* gamma1 = (const float*)d_in[9];
  const float* beta1  = (const float*)d_in[10];
  const float* mean1  = (const float*)d_in[11];
  const float* var1   = (const float*)d_in[12];
  const float* gamma2 = (const float*)d_in[13];
  const float* beta2  = (const float*)d_in[14];
  const float* mean2  = (const float*)d_in[15];
  const float* var2   = (const float*)d_in[16];
  float* out = (float*)d_out;

  const int N = in_sizes[0] / F_IN;        // 50000 (multiple of 16)
  const int E = in_sizes[1] / 2;           // 800000
  const int* src = ei;                     // edge_index[0]
  const int* dst = ei + E;                 // edge_index[1]

  // workspace layout (floats)
  float* ws     = (float*)d_ws;
  float* dinv   = ws;                                  // N      (also deg scratch)
  float* bufHW  = dinv + N;                            // N*64   (GEMM outputs)
  float* bufH   = bufHW + (size_t)N * HID;             // N*64   (agg / activations)
  float* counts = bufH + (size_t)N * HID;              // NGRAPH

  const int TB = 256;
  auto blocks = [](long long n, int tb) { return (unsigned)((n + tb - 1) / tb); };

  // ---- gcn_norm: deg (self-loop => init 1), then rsqrt
  fill_kernel<<<blocks(N, TB), TB, 0, stream>>>(dinv, 1.0f, N);
  deg_accum_kernel<<<blocks(E, TB), TB, 0, stream>>>(dst, dinv, E);
  dinv_kernel<<<blocks(N, TB), TB, 0, stream>>>(dinv, N);

  const int mGroups = (N + 63) / 64;                       // 4 M-tiles per wave
  const int waves64 = mGroups * (HID / 16);
  const int waves32 = mGroups * (EMB / 16);
  const int wavesPerBlk = TB / 32;
  const long long q64 = (long long)N * HID / 4;            // float4 counts
  const long long q32 = (long long)N * EMB / 4;

  // ---- conv1: x @ W1 -> scatter -> BN+ReLU
  gemm_wmma_f32<HID><<<blocks(waves64, wavesPerBlk), TB, 0, stream>>>(x, W1, bufHW, N);
  fill4_kernel<<<blocks(q64, TB), TB, 0, stream>>>((v4f*)bufH, 0.0f, q64);
  scatter_kernel<HID><<<blocks(E, wavesPerBlk), TB, 0, stream>>>(bufHW, src, dst, dinv, bufH, E);
  bn_relu_kernel<<<blocks((long long)N * (HID / 4), TB), TB, 0, stream>>>(
      bufH, bufHW, dinv, b1, gamma1, beta1, mean1, var1, N);

  // ---- conv2
  gemm_wmma_f32<HID><<<blocks(waves64, wavesPerBlk), TB, 0, stream>>>(bufH, W2, bufHW, N);
  fill4_kernel<<<blocks(q64, TB), TB, 0, stream>>>((v4f*)bufH, 0.0f, q64);
  scatter_kernel<HID><<<blocks(E, wavesPerBlk), TB, 0, stream>>>(bufHW, src, dst, dinv, bufH, E);
  bn_relu_kernel<<<blocks((long long)N * (HID / 4), TB), TB, 0, stream>>>(
      bufH, bufHW, dinv, b2, gamma2, beta2, mean2, var2, N);

  // ---- conv3 (HID -> EMB, no BN/ReLU)
  gemm_wmma_f32<EMB><<<blocks(waves32, wavesPerBlk), TB, 0, stream>>>(bufH, W3, bufHW, N);
  fill4_kernel<<<blocks(q32, TB), TB, 0, stream>>>((v4f*)bufH, 0.0f, q32);
  scatter_kernel<EMB><<<blocks(E, wavesPerBlk), TB, 0, stream>>>(bufHW, src, dst, dinv, bufH, E);
  selfloop_bias_kernel<<<blocks((long long)N * (EMB / 4), TB), TB, 0, stream>>>(
      bufH, bufHW, dinv, b3, N);

  // ---- pooling: mean + max per graph
  out_init_kernel<<<blocks(NGRAPH * 2 * EMB, TB), TB, 0, stream>>>(out);
  fill_kernel<<<blocks(NGRAPH, TB), TB, 0, stream>>>(counts, 0.0f, NGRAPH);
  count_kernel<<<blocks(N, TB), TB, 0, stream>>>(batch, counts, N);
  pool_kernel<<<blocks((long long)N * EMB, TB), TB, 0, stream>>>(bufH, batch, out, N);
  finalize_kernel<<<blocks(NGRAPH * EMB, TB), TB, 0, stream>>>(out, counts);
}